// GIN_73718818669208
// MI455X (gfx1250) — compile-verified
//
#include <hip/hip_runtime.h>

#define N_NODES 10000
#define FEATS   128     // IN_FEATS == HIDDEN == 128
#define SCAN_T  256

typedef float v2f __attribute__((ext_vector_type(2)));
typedef float v8f __attribute__((ext_vector_type(8)));

// ---------------------------------------------------------------------------
// Stage A: bucket edges by destination (counting sort, rebuilt every launch).
// These touch only int arrays (40 KB counts, 2.5 MB edge ids) -> negligible.
// ---------------------------------------------------------------------------
__global__ void gin_count_kernel(const int* __restrict__ dst,
                                 int* __restrict__ count, int n_edges) {
    int e = blockIdx.x * blockDim.x + threadIdx.x;
    if (e < n_edges) atomicAdd(&count[dst[e]], 1);
}

// Single-block exclusive scan over N_NODES bins (256 thr x ~40 bins each).
__global__ void gin_scan_kernel(const int* __restrict__ count,
                                int* __restrict__ offsets,
                                int* __restrict__ cursor, int n) {
    __shared__ int partial[SCAN_T];
    const int t   = threadIdx.x;
    const int per = (n + SCAN_T - 1) / SCAN_T;
    const int base = t * per;
    int s = 0;
    for (int i = 0; i < per; ++i) {
        int idx = base + i;
        if (idx < n) s += count[idx];
    }
    partial[t] = s;
    __syncthreads();
    for (int d = 1; d < SCAN_T; d <<= 1) {           // Hillis-Steele inclusive
        int v = (t >= d) ? partial[t - d] : 0;
        __syncthreads();
        partial[t] += v;
        __syncthreads();
    }
    int run = (t > 0) ? partial[t - 1] : 0;          // exclusive prefix
    for (int i = 0; i < per; ++i) {
        int idx = base + i;
        if (idx < n) {
            offsets[idx] = run;
            cursor[idx]  = run;
            run += count[idx];
        }
    }
}

__global__ void gin_bucket_kernel(const int* __restrict__ src,
                                  const int* __restrict__ dst,
                                  int* __restrict__ cursor,
                                  int* __restrict__ sorted_src, int n_edges) {
    int e = blockIdx.x * blockDim.x + threadIdx.x;
    if (e < n_edges) {
        int p = atomicAdd(&cursor[dst[e]], 1);
        sorted_src[p] = src[e];
    }
}

// ---------------------------------------------------------------------------
// Stage B (per layer): pull-based aggregation, NO fp atomics.
//   z[v] = h[v] + sum_{u in in(v)} h[u]        (EPS == 0)
// One wave per node (10000 = 1250 blocks x 8 waves exact). Each lane owns 4
// consecutive feats (float4 -> global_load_b128). 32 edge indices are loaded
// coalesced, then broadcast lane-by-lane via __shfl (ds_bpermute_b32, LDS
// crossbar, no LDS memory traffic). Accumulation stays in VGPRs; all source
// rows are L2-resident (5.1 MB vs 192 MB L2).
// ---------------------------------------------------------------------------
__global__ void gin_gather_z_kernel(const float* __restrict__ h,
                                    const int* __restrict__ sorted_src,
                                    const int* __restrict__ offsets,
                                    const int* __restrict__ count,
                                    float* __restrict__ z) {
    int wave = threadIdx.x >> 5;
    int lane = threadIdx.x & 31;
    int v = blockIdx.x * 8 + wave;

    const int off = offsets[v];
    const int cnt = count[v];

    float4 acc = *reinterpret_cast<const float4*>(h + (size_t)v * FEATS + lane * 4);

    for (int j0 = 0; j0 < cnt; j0 += 32) {
        int valid = cnt - j0;
        if (valid > 32) valid = 32;
        int myIdx = (lane < valid) ? sorted_src[off + j0 + lane] : 0;
        for (int t = 0; t < valid; ++t) {
            int s = __shfl(myIdx, t);                  // wave32 broadcast
            const float4 r = *reinterpret_cast<const float4*>(
                h + (size_t)s * FEATS + lane * 4);
            acc.x += r.x; acc.y += r.y; acc.z += r.z; acc.w += r.w;
        }
    }
    *reinterpret_cast<float4*>(z + (size_t)v * FEATS + lane * 4) = acc;
}

// ---------------------------------------------------------------------------
// Stage C (per layer):  out = act( z @ W + b )  via V_WMMA_F32_16X16X4_F32.
// Exact fp32 matmul on the matrix pipe. Block = 256 thr = 8 waves; block owns
// rows [16b,16b+16), wave w owns cols [16w,16w+16). K=128 -> 32 chained WMMAs.
// M = 10000 = 625*16 exact; EXEC all-ones per participating wave.
// A 16x4 f32 (ISA 7.12.2): lanes 0-15 rows, VGPR0/1 = K0/K1; lanes 16-31 K2/K3.
// ---------------------------------------------------------------------------
__global__ void gin_gemm_wmma_kernel(const float* __restrict__ z,
                                     const float* __restrict__ W,     // K x N
                                     const float* __restrict__ bias,  // N
                                     float*       __restrict__ out,   // M x N
                                     int Ncols, int do_relu) {
    const int K = FEATS;
    int m0   = blockIdx.x * 16;
    int wave = threadIdx.x >> 5;
    int lane = threadIdx.x & 31;
    int n0   = wave * 16;
    if (n0 >= Ncols) return;             // uniform per wave (last layer N=64)

    int half = lane >> 4;                // 0: K pair {0,1}; 1: K pair {2,3}
    int l16  = lane & 15;
    const float* zrow = z + (size_t)(m0 + l16) * K;

    v8f c = {};
#pragma unroll 4
    for (int k = 0; k < K; k += 4) {
        int ka = k + half * 2;
        v2f a, b;
        a.x = zrow[ka + 0];
        a.y = zrow[ka + 1];
        b.x = W[(size_t)(ka + 0) * Ncols + n0 + l16];
        b.y = W[(size_t)(ka + 1) * Ncols + n0 + l16];
        c = __builtin_amdgcn_wmma_f32_16x16x4_f32(
                false, a, false, b, (short)0, c, false, false);
    }

    float bv = bias[n0 + l16];
#pragma unroll
    for (int i = 0; i < 8; ++i) {
        float vv = c[i] + bv;
        if (do_relu) vv = fmaxf(vv, 0.0f);
        out[(size_t)(m0 + i + 8 * half) * Ncols + n0 + l16] = vv;
    }
}

// ---------------------------------------------------------------------------
// 4-layer GIN: bucket once, then (gather_z -> wmma gemm) x 4, ping-pong h.
// ---------------------------------------------------------------------------
extern "C" void kernel_launch(void* const* d_in, const int* in_sizes, int n_in,
                              void* d_out, int out_size, void* d_ws, size_t ws_size,
                              hipStream_t stream) {
    const float* features = (const float*)d_in[0];
    const int*   src      = (const int*)  d_in[1];
    const int*   dst      = (const int*)  d_in[2];
    const float* W1 = (const float*)d_in[3];  const float* b1 = (const float*)d_in[4];
    const float* W3 = (const float*)d_in[5];  const float* b3 = (const float*)d_in[6];
    const float* W4 = (const float*)d_in[7];  const float* b4 = (const float*)d_in[8];
    const float* W2 = (const float*)d_in[9];  const float* b2 = (const float*)d_in[10];

    const int n_edges = in_sizes[1];
    const size_t intBytes  = (size_t)N_NODES * sizeof(int);        // 40 KB (16B-mult)
    const size_t edgeBytes = (size_t)n_edges * sizeof(int);
    const size_t nodeBytes = (size_t)N_NODES * FEATS * sizeof(float); // 5.12 MB

    char* ws = (char*)d_ws;
    int*   count      = (int*)(ws);
    int*   offsets    = (int*)(ws + intBytes);
    int*   cursor     = (int*)(ws + 2 * intBytes);
    int*   sorted_src = (int*)(ws + 3 * intBytes);
    char*  fbase      = ws + 3 * intBytes + ((edgeBytes + 255) & ~(size_t)255);
    float* zbuf = (float*)(fbase);
    float* hA   = (float*)(fbase + nodeBytes);
    float* hB   = (float*)(fbase + 2 * nodeBytes);

    dim3 eblk(256), egrd((n_edges + 255) / 256);
    dim3 gblk(256), ggrd(N_NODES / 8);        // gather: 8 waves/block, exact
    dim3 mblk(256), mgrd(N_NODES / 16);       // gemm: 625 row tiles, exact

    // --- Stage A: bucket edges by destination (once per launch) ---
    hipMemsetAsync(count, 0, intBytes, stream);
    gin_count_kernel <<<egrd, eblk, 0, stream>>>(dst, count, n_edges);
    gin_scan_kernel  <<<1, SCAN_T, 0, stream>>>(count, offsets, cursor, N_NODES);
    gin_bucket_kernel<<<egrd, eblk, 0, stream>>>(src, dst, cursor, sorted_src, n_edges);

    // --- Layer 1: features -> hA ---
    gin_gather_z_kernel<<<ggrd, gblk, 0, stream>>>(features, sorted_src, offsets, count, zbuf);
    gin_gemm_wmma_kernel<<<mgrd, mblk, 0, stream>>>(zbuf, W1, b1, hA, FEATS, 1);

    // --- Layer 2: hA -> hB ---
    gin_gather_z_kernel<<<ggrd, gblk, 0, stream>>>(hA, sorted_src, offsets, count, zbuf);
    gin_gemm_wmma_kernel<<<mgrd, mblk, 0, stream>>>(zbuf, W3, b3, hB, FEATS, 1);

    // --- Layer 3: hB -> hA ---
    gin_gather_z_kernel<<<ggrd, gblk, 0, stream>>>(hB, sorted_src, offsets, count, zbuf);
    gin_gemm_wmma_kernel<<<mgrd, mblk, 0, stream>>>(zbuf, W4, b4, hA, FEATS, 1);

    // --- Layer 4: hA -> d_out (64 classes, no relu) ---
    gin_gather_z_kernel<<<ggrd, gblk, 0, stream>>>(hA, sorted_src, offsets, count, zbuf);
    gin_gemm_wmma_kernel<<<mgrd, mblk, 0, stream>>>(zbuf, W2, b2, (float*)d_out, 64, 0);
}